// blackbox_synthetic_GIN_77171972374915
// MI455X (gfx1250) — compile-verified
//
#include <hip/hip_runtime.h>
#include <hip/hip_bf16.h>

typedef __attribute__((ext_vector_type(2))) float v2f;
typedef __attribute__((ext_vector_type(8))) float v8f;

#define N_NODES 50000
#define N_EDGES 800000
#define HID 64
#define N_CLASSES 10

// ---------------------------------------------------------------------------
// native fp32 global atomic add (no return) -> STOREcnt, completes by endpgm
// ---------------------------------------------------------------------------
__device__ __forceinline__ void atomic_add_f32_native(float* p, float v) {
    asm volatile("global_atomic_add_f32 %0, %1, off" :: "v"(p), "v"(v) : "memory");
}

// ---------------------------------------------------------------------------
// h = x  (vectorized copy; h then accumulates the neighbor sum -> h = x + agg)
// ---------------------------------------------------------------------------
__global__ __launch_bounds__(256) void copy_f4_kernel(const float* __restrict__ src,
                                                      float* __restrict__ dst, int n4) {
    int i = blockIdx.x * 256 + threadIdx.x;
    if (i < n4) ((float4*)dst)[i] = ((const float4*)src)[i];
}

// ---------------------------------------------------------------------------
// edge aggregation: 16 threads per edge, float4 gather + 4 fp32 atomics
// ---------------------------------------------------------------------------
__global__ __launch_bounds__(256) void edge_agg_kernel(const float* __restrict__ x,
                                                       const long long* __restrict__ src,
                                                       const long long* __restrict__ dst,
                                                       float* __restrict__ h) {
    long long tid = (long long)blockIdx.x * 256 + threadIdx.x;
    long long e = tid >> 4;
    int c = (int)(tid & 15);
    if (e >= N_EDGES) return;
    long long s = src[e];
    long long d = dst[e];
    float4 v = *(const float4*)(x + s * HID + c * 4);
    float* o = h + d * HID + c * 4;
    atomic_add_f32_native(o + 0, v.x);
    atomic_add_f32_native(o + 1, v.y);
    atomic_add_f32_native(o + 2, v.z);
    atomic_add_f32_native(o + 3, v.w);
}

// ---------------------------------------------------------------------------
// out = relu( relu(h @ W1 + b1) @ W2 + b2 )   [fused GIN MLP + outer relu]
// block = 256 threads = 8 waves; each wave owns 16 rows; WMMA f32 16x16x4.
// A layout: lane%16 = M row; lane/16 selects K pair {0,1}/{2,3} within K-quad.
// C/D layout: vgpr r -> M = r + (lane/16)*8, N = lane%16 (+ 16*nt).
// ---------------------------------------------------------------------------
__global__ __launch_bounds__(256) void gin_mlp_kernel(const float* __restrict__ h,
                                                      const float* __restrict__ W1,
                                                      const float* __restrict__ b1,
                                                      const float* __restrict__ W2,
                                                      const float* __restrict__ b2,
                                                      float* __restrict__ out) {
    __shared__ float sW1[HID * HID];
    __shared__ float sW2[HID * HID];
    __shared__ float sB1[HID];
    __shared__ float sB2[HID];
    __shared__ float sHid[8][16 * HID];   // per-wave re-layout staging (32 KB)

    const int t = threadIdx.x;
    for (int i = t; i < HID * HID; i += 256) { sW1[i] = W1[i]; sW2[i] = W2[i]; }
    if (t < HID) { sB1[t] = b1[t]; sB2[t] = b2[t]; }
    __syncthreads();

    const int wave  = t >> 5;
    const int lane  = t & 31;
    const int lrow  = lane & 15;   // M (or N) index within tile
    const int khalf = lane >> 4;   // selects K pair within each K-quad

    const int rowBase = blockIdx.x * 128 + wave * 16;
    int row = rowBase + lrow;
    if (row >= N_NODES) row = N_NODES - 1;   // clamp loads; stores masked below

    // ---- load A fragments of h (16 rows x 64 K) ----
    v2f a[16];
    const float* hrow = h + (long long)row * HID + khalf * 2;
#pragma unroll
    for (int kb = 0; kb < 16; ++kb) {
        float2 p = *(const float2*)(hrow + 4 * kb);
        a[kb].x = p.x; a[kb].y = p.y;
    }

    float* myHid = &sHid[wave][0];

    // ---- hidden = relu(h @ W1 + b1), staged to LDS in A-friendly layout ----
#pragma unroll
    for (int nt = 0; nt < 4; ++nt) {
        const int n = nt * 16 + lrow;
        const float bv = sB1[n];
        v8f c;
#pragma unroll
        for (int r = 0; r < 8; ++r) c[r] = bv;
#pragma unroll
        for (int kb = 0; kb < 16; ++kb) {
            const int kbase = 4 * kb + khalf * 2;
            v2f b;
            b.x = sW1[kbase * HID + n];
            b.y = sW1[(kbase + 1) * HID + n];
            c = __builtin_amdgcn_wmma_f32_16x16x4_f32(false, a[kb], false, b,
                                                      (short)0, c, false, false);
        }
#pragma unroll
        for (int r = 0; r < 8; ++r) {
            float v = c[r] > 0.0f ? c[r] : 0.0f;
            myHid[(r + khalf * 8) * HID + n] = v;     // row-major 16x64 tile
        }
    }
    __syncthreads();

    // ---- reload hidden as A fragments ----
    v2f a2[16];
    const float* hidrow = myHid + lrow * HID + khalf * 2;
#pragma unroll
    for (int kb = 0; kb < 16; ++kb) {
        a2[kb].x = hidrow[4 * kb];
        a2[kb].y = hidrow[4 * kb + 1];
    }

    // ---- out = relu(hidden @ W2 + b2) ----
#pragma unroll
    for (int nt = 0; nt < 4; ++nt) {
        const int n = nt * 16 + lrow;
        const float bv = sB2[n];
        v8f c;
#pragma unroll
        for (int r = 0; r < 8; ++r) c[r] = bv;
#pragma unroll
        for (int kb = 0; kb < 16; ++kb) {
            const int kbase = 4 * kb + khalf * 2;
            v2f b;
            b.x = sW2[kbase * HID + n];
            b.y = sW2[(kbase + 1) * HID + n];
            c = __builtin_amdgcn_wmma_f32_16x16x4_f32(false, a2[kb], false, b,
                                                      (short)0, c, false, false);
        }
#pragma unroll
        for (int r = 0; r < 8; ++r) {
            const int m = rowBase + r + khalf * 8;
            if (m < N_NODES) {
                float v = c[r] > 0.0f ? c[r] : 0.0f;
                out[(long long)m * HID + n] = v;
            }
        }
    }
}

// ---------------------------------------------------------------------------
// out = x @ W_lin + b_lin   (64 -> 10, W padded to 64x16 with zeros in LDS)
// ---------------------------------------------------------------------------
__global__ __launch_bounds__(256) void final_linear_kernel(const float* __restrict__ x,
                                                           const float* __restrict__ Wl,
                                                           const float* __restrict__ bl,
                                                           float* __restrict__ out) {
    __shared__ float sW[HID * 16];
    __shared__ float sB[16];
    const int t = threadIdx.x;
    for (int i = t; i < HID * 16; i += 256) {
        int k = i >> 4, n = i & 15;
        sW[i] = (n < N_CLASSES) ? Wl[k * N_CLASSES + n] : 0.0f;
    }
    if (t < 16) sB[t] = (t < N_CLASSES) ? bl[t] : 0.0f;
    __syncthreads();

    const int wave  = t >> 5;
    const int lane  = t & 31;
    const int lrow  = lane & 15;
    const int khalf = lane >> 4;

    const int rowBase = blockIdx.x * 128 + wave * 16;
    int row = rowBase + lrow;
    if (row >= N_NODES) row = N_NODES - 1;

    v2f a[16];
    const float* xrow = x + (long long)row * HID + khalf * 2;
#pragma unroll
    for (int kb = 0; kb < 16; ++kb) {
        float2 p = *(const float2*)(xrow + 4 * kb);
        a[kb].x = p.x; a[kb].y = p.y;
    }

    const float bv = sB[lrow];
    v8f c;
#pragma unroll
    for (int r = 0; r < 8; ++r) c[r] = bv;
#pragma unroll
    for (int kb = 0; kb < 16; ++kb) {
        const int kbase = 4 * kb + khalf * 2;
        v2f b;
        b.x = sW[kbase * 16 + lrow];
        b.y = sW[(kbase + 1) * 16 + lrow];
        c = __builtin_amdgcn_wmma_f32_16x16x4_f32(false, a[kb], false, b,
                                                  (short)0, c, false, false);
    }
#pragma unroll
    for (int r = 0; r < 8; ++r) {
        const int m = rowBase + r + khalf * 8;
        if (m < N_NODES && lrow < N_CLASSES)
            out[(long long)m * N_CLASSES + lrow] = c[r];
    }
}

// ---------------------------------------------------------------------------
extern "C" void kernel_launch(void* const* d_in, const int* in_sizes, int n_in,
                              void* d_out, int out_size, void* d_ws, size_t ws_size,
                              hipStream_t stream) {
    const float* feats      = (const float*)d_in[0];
    const long long* edges  = (const long long*)d_in[1];   // int64 (2, N_EDGES)
    const long long* src    = edges;
    const long long* dst    = edges + N_EDGES;

    const float* W1[3] = { (const float*)d_in[2],  (const float*)d_in[6],  (const float*)d_in[10] };
    const float* B1[3] = { (const float*)d_in[3],  (const float*)d_in[7],  (const float*)d_in[11] };
    const float* W2[3] = { (const float*)d_in[4],  (const float*)d_in[8],  (const float*)d_in[12] };
    const float* B2[3] = { (const float*)d_in[5],  (const float*)d_in[9],  (const float*)d_in[13] };
    const float* Wlin  = (const float*)d_in[14];
    const float* blin  = (const float*)d_in[15];

    float* hbuf = (float*)d_ws;                               // 50000*64 f32
    float* xbuf = hbuf + (size_t)N_NODES * HID;               // 50000*64 f32
    float* out  = (float*)d_out;

    const int NF4        = N_NODES * HID / 4;                 // 800000 float4
    const int copyBlocks = (NF4 + 255) / 256;
    const int edgeBlocks = (int)(((long long)N_EDGES * 16 + 255) / 256);
    const int mlpBlocks  = (N_NODES + 127) / 128;             // 391

    for (int l = 0; l < 3; ++l) {
        const float* xin = (l == 0) ? feats : xbuf;
        copy_f4_kernel<<<copyBlocks, 256, 0, stream>>>(xin, hbuf, NF4);
        edge_agg_kernel<<<edgeBlocks, 256, 0, stream>>>(xin, src, dst, hbuf);
        gin_mlp_kernel<<<mlpBlocks, 256, 0, stream>>>(hbuf, W1[l], B1[l], W2[l], B2[l], xbuf);
    }
    final_linear_kernel<<<mlpBlocks, 256, 0, stream>>>(xbuf, Wlin, blin, out);
}